// SSDBoxHead_70901320122849
// MI455X (gfx1250) — compile-verified
//
#include <hip/hip_runtime.h>
#include <stdint.h>

#define BATCH      32
#define NPRIORS    8732
#define NCLASSES   81
#define NCLS       80          // foreground classes
#define TOPK       100
#define MAXDET     100
#define CENTER_VAR 0.1f
#define SIZE_VAR   0.2f
#define CONF_THR   0.01f
#define NMS_THR    0.45f
#define NCHUNK     21          // ceil(81/4) K-chunks of 4 for wmma 16x16x4
#define NEG_BIG    -1.0e30f

typedef __attribute__((ext_vector_type(2))) float        v2f;
typedef __attribute__((ext_vector_type(8))) float        v8f;
typedef __attribute__((ext_vector_type(4))) unsigned int u32x4;
typedef __attribute__((ext_vector_type(8))) int          i32x8;
typedef __attribute__((ext_vector_type(4))) int          i32x4;

// ---------------------------------------------------------------------------
// Kernel 1: softmax over 81 classes, row-sums via v_wmma_f32_16x16x4_f32.
// One wave handles 16 priors. A = exp-chunk [16 priors x 4 classes],
// B = ones [4 x 16]  =>  D[i][j] = row_sum(i) replicated across columns.
// Writes scores transposed [B][NCLS][NPRIORS] (background class dropped).
// ---------------------------------------------------------------------------
__global__ __launch_bounds__(256)
void softmax_wmma_kernel(const float* __restrict__ logits,
                         float* __restrict__ scores_t) {
  const int b    = blockIdx.y;
  const int wave = threadIdx.x >> 5;
  const int lane = threadIdx.x & 31;
  const int row  = lane & 15;     // prior within wave tile (M index)
  const int half = lane >> 4;     // 0 -> cols {4c,4c+1}, 1 -> cols {4c+2,4c+3}
  const int prior = blockIdx.x * 128 + wave * 16 + row;
  const int p     = (prior < NPRIORS) ? prior : (NPRIORS - 1);
  const bool valid = (prior < NPRIORS);

  const float* __restrict__ src = logits + ((size_t)b * NPRIORS + p) * NCLASSES;

  float e0[NCHUNK], e1[NCHUNK];
  float m = NEG_BIG;
#pragma unroll
  for (int c = 0; c < NCHUNK; ++c) {
    const int col = 4 * c + 2 * half;
    const float l0 = (col     < NCLASSES) ? src[col]     : NEG_BIG;
    const float l1 = (col + 1 < NCLASSES) ? src[col + 1] : NEG_BIG;
    e0[c] = l0; e1[c] = l1;
    m = fmaxf(m, fmaxf(l0, l1));
  }
  // merge the two half-lane partial maxes of this prior's row
  m = fmaxf(m, __shfl_xor(m, 16, 32));

  v8f acc = {0.f, 0.f, 0.f, 0.f, 0.f, 0.f, 0.f, 0.f};
  const v2f ones = {1.f, 1.f};
#pragma unroll
  for (int c = 0; c < NCHUNK; ++c) {
    const int col = 4 * c + 2 * half;
    const float a0 = (col     < NCLASSES) ? __expf(e0[c] - m) : 0.f;
    const float a1 = (col + 1 < NCLASSES) ? __expf(e1[c] - m) : 0.f;
    e0[c] = a0; e1[c] = a1;
    v2f a = {a0, a1};
    // D = A(16x4) * ones(4x16) + C  -> every column of row i = row_sum(i)
    acc = __builtin_amdgcn_wmma_f32_16x16x4_f32(false, a, false, ones,
                                                (short)0, acc, false, false);
  }

  // D layout: VGPR r holds M=r (lanes 0-15) and M=r+8 (lanes 16-31).
  __shared__ float sums[8][16];
  if (lane == 0) {
#pragma unroll
    for (int r = 0; r < 8; ++r) sums[wave][r] = acc[r];
  }
  if (lane == 16) {
#pragma unroll
    for (int r = 0; r < 8; ++r) sums[wave][8 + r] = acc[r];
  }
  __syncthreads();

  const float inv = 1.0f / sums[wave][row];
  if (valid) {
    float* __restrict__ dstb = scores_t + (size_t)b * NCLS * NPRIORS;
#pragma unroll
    for (int c = 0; c < NCHUNK; ++c) {
      const int col = 4 * c + 2 * half;
      if (col >= 1 && col < NCLASSES)
        dstb[(size_t)(col - 1) * NPRIORS + prior] = e0[c] * inv;
      if (col + 1 >= 1 && col + 1 < NCLASSES)
        dstb[(size_t)(col) * NPRIORS + prior] = e1[c] * inv;
    }
  }
}

// ---------------------------------------------------------------------------
// Kernel 2: box decode (center-form priors -> corner-form boxes)
// ---------------------------------------------------------------------------
__global__ __launch_bounds__(256)
void decode_kernel(const float* __restrict__ loc,
                   const float* __restrict__ priors,
                   float* __restrict__ boxes) {
  const int tid = blockIdx.x * blockDim.x + threadIdx.x;
  if (tid >= BATCH * NPRIORS) return;
  const int n = tid % NPRIORS;
  const float l0 = loc[(size_t)tid * 4 + 0];
  const float l1 = loc[(size_t)tid * 4 + 1];
  const float l2 = loc[(size_t)tid * 4 + 2];
  const float l3 = loc[(size_t)tid * 4 + 3];
  const float px = priors[(size_t)n * 4 + 0];
  const float py = priors[(size_t)n * 4 + 1];
  const float pw = priors[(size_t)n * 4 + 2];
  const float ph = priors[(size_t)n * 4 + 3];
  const float cx = l0 * CENTER_VAR * pw + px;
  const float cy = l1 * CENTER_VAR * ph + py;
  const float w  = __expf(l2 * SIZE_VAR) * pw;
  const float h  = __expf(l3 * SIZE_VAR) * ph;
  boxes[(size_t)tid * 4 + 0] = cx - 0.5f * w;
  boxes[(size_t)tid * 4 + 1] = cy - 0.5f * h;
  boxes[(size_t)tid * 4 + 2] = cx + 0.5f * w;
  boxes[(size_t)tid * 4 + 3] = cy + 0.5f * h;
}

// ---------------------------------------------------------------------------
// Kernel 3: per-(batch,class) top-100 + greedy NMS.
// Score row staged global->LDS with the Tensor Data Mover (TDM).
// grid = (NCLS, BATCH), 256 threads.
// ---------------------------------------------------------------------------
__global__ __launch_bounds__(256)
void nms_kernel(const float* __restrict__ scores_t,
                const float* __restrict__ boxes,
                float* __restrict__ s_k,
                float* __restrict__ b_k) {
  const int c = blockIdx.x;           // foreground class index 0..79
  const int b = blockIdx.y;
  const int t = threadIdx.x;

  __shared__ float sc[NPRIORS];       // 34928 B
  __shared__ float rv[256];
  __shared__ int   ri[256];
  __shared__ float sel_s[TOPK];
  __shared__ int   sel_i[TOPK];
  __shared__ float bx1[TOPK], by1[TOPK], bx2[TOPK], by2[TOPK], area[TOPK];
  __shared__ int   keep[TOPK];

  const float* __restrict__ srow = scores_t + ((size_t)b * NCLS + c) * NPRIORS;

  // --- TDM: DMA the whole 8732-float score row into LDS (wave 0 issues) ---
  if ((threadIdx.x >> 5) == 0) {
    const unsigned lds_addr = (unsigned)(size_t)(&sc[0]);
    const unsigned long long ga = (unsigned long long)(size_t)srow;
    u32x4 g0;
    g0[0] = 1u;                                   // count=1, user descriptor
    g0[1] = lds_addr;                             // lds_addr[31:0]
    g0[2] = (unsigned)(ga & 0xFFFFFFFFu);         // global_addr[31:0]
    g0[3] = (unsigned)((ga >> 32) & 0x1FFFFFFu)   // global_addr[56:32]
          | (2u << 30);                           // type=2 ("image")
    i32x8 g1;
    g1[0] = 0x20000;                              // data_size=2 (4 bytes)
    g1[1] = (int)((NPRIORS & 0xFFFF) << 16);      // tensor_dim0 lo16
    g1[2] = (int)((NPRIORS >> 16) | (1 << 16));   // tensor_dim0 hi16 | tensor_dim1=1
    g1[3] = (int)(NPRIORS << 16);                 // tensor_dim1 hi | tile_dim0
    g1[4] = 1;                                    // tile_dim1=1, tile_dim2=0
    g1[5] = NPRIORS;                              // tensor_dim0_stride lo32
    g1[6] = 0;                                    // stride hi | dim1_stride lo
    g1[7] = 0;
    i32x4 g2 = {0, 0, 0, 0};
    i32x4 g3 = {0, 0, 0, 0};
    i32x8 gz = {0, 0, 0, 0, 0, 0, 0, 0};          // extra group (6-arg builtin)
    __builtin_amdgcn_tensor_load_to_lds(g0, g1, g2, g3, gz, 0);
    __builtin_amdgcn_s_wait_tensorcnt(0);
  }
  __syncthreads();

  // --- iterative top-100 (stable: ties -> lowest index, like lax.top_k) ---
  for (int k = 0; k < TOPK; ++k) {
    float bv = NEG_BIG;
    int   bi = NPRIORS;
    for (int i = t; i < NPRIORS; i += 256) {
      const float v = sc[i];
      if (v > bv) { bv = v; bi = i; }
    }
    rv[t] = bv; ri[t] = bi;
    __syncthreads();
    for (int s = 128; s > 0; s >>= 1) {
      if (t < s) {
        const float ov = rv[t + s]; const int oi = ri[t + s];
        if (ov > rv[t] || (ov == rv[t] && oi < ri[t])) { rv[t] = ov; ri[t] = oi; }
      }
      __syncthreads();
    }
    if (t == 0) {
      sel_s[k] = rv[0];
      sel_i[k] = ri[0];
      sc[ri[0]] = NEG_BIG;                        // remove winner
    }
    __syncthreads();
  }

  // --- gather boxes, init keep ---
  if (t < TOPK) {
    const float* bb = boxes + ((size_t)b * NPRIORS + sel_i[t]) * 4;
    const float x1 = bb[0], y1 = bb[1], x2 = bb[2], y2 = bb[3];
    bx1[t] = x1; by1[t] = y1; bx2[t] = x2; by2[t] = y2;
    area[t] = fmaxf(x2 - x1, 0.f) * fmaxf(y2 - y1, 0.f);
    keep[t] = (sel_s[t] > CONF_THR) ? 1 : 0;
  }
  __syncthreads();

  // --- greedy NMS on sorted candidates ---
  for (int i = 0; i < TOPK; ++i) {
    __syncthreads();
    if (t < TOPK && t > i && keep[i] && keep[t]) {
      const float lx = fmaxf(bx1[i], bx1[t]);
      const float ly = fmaxf(by1[i], by1[t]);
      const float rx = fminf(bx2[i], bx2[t]);
      const float ry = fminf(by2[i], by2[t]);
      const float iw = fmaxf(rx - lx, 0.f);
      const float ih = fmaxf(ry - ly, 0.f);
      const float inter = iw * ih;
      const float iou = inter / (area[i] + area[t] - inter + 1e-9f);
      if (iou > NMS_THR) keep[t] = 0;
    }
  }
  __syncthreads();

  if (t < TOPK) {
    const size_t o = ((size_t)b * NCLS + c) * TOPK + t;
    s_k[o] = keep[t] ? sel_s[t] : 0.f;
    float* bo = b_k + o * 4;
    bo[0] = bx1[t]; bo[1] = by1[t]; bo[2] = bx2[t]; bo[3] = by2[t];
  }
}

// ---------------------------------------------------------------------------
// Kernel 4: per-image final top-100 over 80*100 candidates.
// ---------------------------------------------------------------------------
#define FLATN (NCLS * TOPK)    // 8000

__global__ __launch_bounds__(256)
void final_topk_kernel(const float* __restrict__ s_k,
                       const float* __restrict__ b_k,
                       float* __restrict__ out_b,
                       float* __restrict__ out_s,
                       int* __restrict__ out_l) {
  const int b = blockIdx.x;
  const int t = threadIdx.x;

  __shared__ float sc[FLATN];      // 32 KB
  __shared__ float rv[256];
  __shared__ int   ri[256];
  __shared__ float sel_s[MAXDET];
  __shared__ int   sel_i[MAXDET];

  const float* __restrict__ srow = s_k + (size_t)b * FLATN;
  for (int i = t; i < FLATN; i += 256) sc[i] = srow[i];
  __syncthreads();

  for (int k = 0; k < MAXDET; ++k) {
    float bv = NEG_BIG;
    int   bi = FLATN;
    for (int i = t; i < FLATN; i += 256) {
      const float v = sc[i];
      if (v > bv) { bv = v; bi = i; }
    }
    rv[t] = bv; ri[t] = bi;
    __syncthreads();
    for (int s = 128; s > 0; s >>= 1) {
      if (t < s) {
        const float ov = rv[t + s]; const int oi = ri[t + s];
        if (ov > rv[t] || (ov == rv[t] && oi < ri[t])) { rv[t] = ov; ri[t] = oi; }
      }
      __syncthreads();
    }
    if (t == 0) {
      sel_s[k] = rv[0];
      sel_i[k] = ri[0];
      sc[ri[0]] = NEG_BIG;
    }
    __syncthreads();
  }

  if (t < MAXDET) {
    const int f   = sel_i[t];
    const int cls = f / TOPK;
    const int kk  = f - cls * TOPK;
    out_s[(size_t)b * MAXDET + t] = sel_s[t];
    out_l[(size_t)b * MAXDET + t] = cls + 1;
    const float* bb = b_k + (((size_t)b * NCLS + cls) * TOPK + kk) * 4;
    float* bo = out_b + ((size_t)b * MAXDET + t) * 4;
    bo[0] = bb[0]; bo[1] = bb[1]; bo[2] = bb[2]; bo[3] = bb[3];
  }
}

// ---------------------------------------------------------------------------
extern "C" void kernel_launch(void* const* d_in, const int* in_sizes, int n_in,
                              void* d_out, int out_size, void* d_ws, size_t ws_size,
                              hipStream_t stream) {
  const float* cls_logits = (const float*)d_in[0];   // [32, 8732, 81]
  const float* bbox_pred  = (const float*)d_in[1];   // [32, 8732, 4]
  const float* priors     = (const float*)d_in[2];   // [8732, 4]

  // workspace layout (floats)
  float* scores_t = (float*)d_ws;                                  // [B][80][N]
  float* boxes    = scores_t + (size_t)BATCH * NCLS * NPRIORS;     // [B][N][4]
  float* s_k      = boxes    + (size_t)BATCH * NPRIORS * 4;        // [B][80][100]
  float* b_k      = s_k      + (size_t)BATCH * NCLS * TOPK;        // [B][80][100][4]

  // output layout: top_b [32,100,4] | top_s [32,100] | top_l [32,100] (int32 bits)
  float* out   = (float*)d_out;
  float* out_b = out;
  float* out_s = out + (size_t)BATCH * MAXDET * 4;
  int*   out_l = (int*)(out + (size_t)BATCH * MAXDET * 5);

  // 1) softmax (WMMA row-sums) -> transposed scores
  {
    dim3 grid((NPRIORS + 127) / 128, BATCH);
    softmax_wmma_kernel<<<grid, 256, 0, stream>>>(cls_logits, scores_t);
  }
  // 2) decode boxes
  {
    const int n = BATCH * NPRIORS;
    decode_kernel<<<(n + 255) / 256, 256, 0, stream>>>(bbox_pred, priors, boxes);
  }
  // 3) per-class top-100 + NMS (TDM stages score rows into LDS)
  {
    dim3 grid(NCLS, BATCH);
    nms_kernel<<<grid, 256, 0, stream>>>(scores_t, boxes, s_k, b_k);
  }
  // 4) final per-image top-100
  {
    final_topk_kernel<<<BATCH, 256, 0, stream>>>(s_k, b_k, out_b, out_s, out_l);
  }
}